// Gate_81217831567442
// MI455X (gfx1250) — compile-verified
//
#include <hip/hip_runtime.h>

// ---------------- problem constants (match reference) ----------------
#define K_TOP   2
#define ALPHA_C 0.01f
#define BETA_C  0.1f
#define D_DIM   2048
#define E_DIM   64
#define NTOK    16384                      // B*T
#define TOK_PER_WAVE 32                    // two 16-row WMMA tiles per wave
#define WAVES_PER_WG 8
#define TOK_PER_WG   (WAVES_PER_WG * TOK_PER_WAVE)   // 256
#define NWG          (NTOK / TOK_PER_WG)             // 64
#define KC           256                   // K-chunk staged in LDS
#define KSTEPS       (KC / 4)              // 64 WMMA k-steps per chunk
#define NCHUNK       (D_DIM / KC)          // 8
#define LOGIT_PITCH  65                    // 64 + 1 pad (conflict-free token reads)

typedef float v2f __attribute__((ext_vector_type(2)));
typedef float v8f __attribute__((ext_vector_type(8)));

// ---------------------------------------------------------------------
// Stage 1: 256 tokens/WG.  Each wave: 32 tokens x 64 experts via
// V_WMMA_F32_16X16X4_F32.  B fragments (identical for every wave) are
// cooperatively staged pre-swizzled in LDS once per K-chunk.
// ---------------------------------------------------------------------
__global__ __launch_bounds__(256) void router_gemm_topk(
    const float* __restrict__ x,      // [NTOK, D]
    const float* __restrict__ W,      // [D, E]
    const float* __restrict__ bias,   // [E]
    float* __restrict__ out,          // [2*NTOK*K + 2]
    float* __restrict__ wsF,          // [NWG, E]
    float* __restrict__ wsP,          // [NWG, E]
    float* __restrict__ wsZ)          // [NWG]
{
    // 16640 floats = 66,560 B.  During GEMM: B-fragment buffer
    // (KSTEPS*4*32 float2 = 64 KB).  After GEMM: logits/scores [256][65].
    __shared__ __align__(16) float smem[TOK_PER_WG * LOGIT_PITCH];
    __shared__ float lds_maxs[TOK_PER_WG];
    __shared__ float lds_lse2[TOK_PER_WG];

    const int tid  = threadIdx.x;
    const int wave = tid >> 5;
    const int lane = tid & 31;
    const int half = lane >> 4;     // A/B fragment K-half (lanes 16-31 -> K+2)
    const int lr   = lane & 15;

    const int wg      = blockIdx.x;
    const int tokBase = wg * TOK_PER_WG + wave * TOK_PER_WAVE;

    v8f acc[8];                      // [rowTile(2)][expertTile(4)]
#pragma unroll
    for (int i = 0; i < 8; ++i) acc[i] = (v8f){};

    const float* xrow0 = x + (size_t)(tokBase + lr) * D_DIM;
    const float* xrow1 = xrow0 + (size_t)16 * D_DIM;

    for (int c = 0; c < NCHUNK; ++c) {
        const int kbase = c * KC;
        __syncthreads();             // previous chunk's fragment reads done

        // ---- stage B fragments, pre-swizzled: entry idx = s*128 + t*32 + l ----
        // entry(l) = { W[ka][e], W[ka+1][e] },  ka = kbase+4s+2*(l>>4), e = 16t+(l&15)
#pragma unroll
        for (int j = 0; j < 32; ++j) {
            const int idx = j * 256 + tid;
            const int l = idx & 31;
            const int t = (idx >> 5) & 3;
            const int s = idx >> 7;
            const int ka = kbase + 4 * s + 2 * (l >> 4);
            const int e  = t * 16 + (l & 15);
            v2f w;
            w.x = W[(size_t)ka * E_DIM + e];
            w.y = W[(size_t)(ka + 1) * E_DIM + e];
            *(v2f*)(smem + ((size_t)idx << 1)) = w;
        }
        __syncthreads();

        if (c + 1 < NCHUNK) {        // keep the x HBM stream ahead of the WMMAs
            __builtin_prefetch(xrow0 + kbase + KC, 0, 1);
            __builtin_prefetch(xrow1 + kbase + KC, 0, 1);
        }

        // ---- inner GEMM: 2 global b64 (A) + 4 ds b64 (B) + 8 WMMA per k-step ----
#pragma unroll 4
        for (int s = 0; s < KSTEPS; ++s) {
            const int ka = kbase + 4 * s + 2 * half;
            v2f a0 = *(const v2f*)(xrow0 + ka);
            v2f a1 = *(const v2f*)(xrow1 + ka);
#pragma unroll
            for (int t = 0; t < 4; ++t) {
                v2f b = *(const v2f*)(smem + (((s * 4 + t) * 32 + lane) << 1));
                acc[t]     = __builtin_amdgcn_wmma_f32_16x16x4_f32(false, a0, false, b, (short)0, acc[t],     false, false);
                acc[4 + t] = __builtin_amdgcn_wmma_f32_16x16x4_f32(false, a1, false, b, (short)0, acc[4 + t], false, false);
            }
        }
    }
    __syncthreads();                 // fragment buffer dead; reuse as logits

    // ---- bias + spill logits.  C/D layout: VGPR r -> M = r + 8*half, N = lr ----
    const float be0 = bias[ 0 + lr];
    const float be1 = bias[16 + lr];
    const float be2 = bias[32 + lr];
    const float be3 = bias[48 + lr];
#pragma unroll
    for (int rt = 0; rt < 2; ++rt) {
#pragma unroll
        for (int r = 0; r < 8; ++r) {
            const int m = wave * TOK_PER_WAVE + rt * 16 + r + 8 * half;
            float* row = smem + m * LOGIT_PITCH;
            row[ 0 + lr] = acc[rt * 4 + 0][r] + be0;
            row[16 + lr] = acc[rt * 4 + 1][r] + be1;
            row[32 + lr] = acc[rt * 4 + 2][r] + be2;
            row[48 + lr] = acc[rt * 4 + 3][r] + be3;
        }
    }
    __syncthreads();

    // ---------------- per-token softmax / top-2 / combine ----------------
    {
        float* lrow = smem + tid * LOGIT_PITCH;

        // stable top-2 of logits (ties -> lower index, like jax.lax.top_k)
        float v1 = -__builtin_inff(), v2 = -__builtin_inff();
        int   i1 = 0, i2 = 0;
        for (int e = 0; e < E_DIM; ++e) {
            const float l = lrow[e];
            if (l > v1)      { v2 = v1; i2 = i1; v1 = l; i1 = e; }
            else if (l > v2) { v2 = l;  i2 = e; }
        }

        float sumexp = 0.f;
        for (int e = 0; e < E_DIM; ++e) sumexp += __expf(lrow[e] - v1);
        const float inv  = 1.0f / sumexp;    // == max softmax score
        const float smax = inv;

        // scores overwrite logits in LDS; accumulate z-loss lse of *scores*
        float zs = 0.f;
        for (int e = 0; e < E_DIM; ++e) {
            const float s = __expf(lrow[e] - v1) * inv;
            zs += __expf(s - smax);
            lrow[e] = s;
        }
        const float lse = smax + __logf(zs);
        lds_lse2[tid] = lse * lse;
        lds_maxs[tid] = smax;

        const int token = wg * TOK_PER_WG + tid;
        const float e2   = __expf(v2 - v1);  // top-1 exp == 1
        const float csum = 1.0f + e2;
        out[token * K_TOP + 0] = (float)i1;
        out[token * K_TOP + 1] = (float)i2;
        out[NTOK * K_TOP + token * K_TOP + 0] = 1.0f / csum;
        out[NTOK * K_TOP + token * K_TOP + 1] = e2 / csum;
    }
    __syncthreads();

    // ---------------- deterministic per-expert partials ----------------
    if (tid < E_DIM) {
        float fsum = 0.f, psum = 0.f;
        for (int t = 0; t < TOK_PER_WG; ++t) {
            const float s = smem[t * LOGIT_PITCH + tid];
            psum += s;
            fsum += (s == lds_maxs[t]) ? 1.0f : 0.0f;
        }
        wsF[wg * E_DIM + tid] = fsum;
        wsP[wg * E_DIM + tid] = psum;
    }
    if (tid == 0) {
        float z = 0.f;
        for (int t = 0; t < TOK_PER_WG; ++t) z += lds_lse2[t];
        wsZ[wg] = z;
    }
}

// ---------------------------------------------------------------------
// Stage 2: fold NWG partials into the two scalar losses.
// ---------------------------------------------------------------------
__global__ __launch_bounds__(64) void router_losses(
    const float* __restrict__ wsF, const float* __restrict__ wsP,
    const float* __restrict__ wsZ, float* __restrict__ out)
{
    __shared__ float red[E_DIM];
    const int e = threadIdx.x;

    float fsum = 0.f, psum = 0.f;
    for (int wg = 0; wg < NWG; ++wg) {
        fsum += wsF[wg * E_DIM + e];
        psum += wsP[wg * E_DIM + e];
    }
    const float f = fsum / (float)NTOK;
    const float P = psum / (float)NTOK;
    red[e] = f * P;
    __syncthreads();

    if (e == 0) {
        float s = 0.f;
        for (int i = 0; i < E_DIM; ++i) s += red[i];
        float z = 0.f;
        for (int wg = 0; wg < NWG; ++wg) z += wsZ[wg];
        out[NTOK * 2 * K_TOP + 0] = ALPHA_C * (s / (float)E_DIM);
        out[NTOK * 2 * K_TOP + 1] = BETA_C  * (z / (float)NTOK);
    }
}

// ---------------------------------------------------------------------
extern "C" void kernel_launch(void* const* d_in, const int* in_sizes, int n_in,
                              void* d_out, int out_size, void* d_ws, size_t ws_size,
                              hipStream_t stream)
{
    const float* x    = (const float*)d_in[0];
    const float* W    = (const float*)d_in[1];
    const float* bias = (const float*)d_in[2];
    float*       out  = (float*)d_out;

    float* wsF = (float*)d_ws;
    float* wsP = wsF + (size_t)NWG * E_DIM;
    float* wsZ = wsP + (size_t)NWG * E_DIM;

    router_gemm_topk<<<NWG, 256, 0, stream>>>(x, W, bias, out, wsF, wsP, wsZ);
    router_losses<<<1, 64, 0, stream>>>(wsF, wsP, wsZ, out);
}